// BiaffineEdgeAttention_56049323213772
// MI455X (gfx1250) — compile-verified
//
#include <hip/hip_runtime.h>

typedef __attribute__((ext_vector_type(2))) float v2f;
typedef __attribute__((ext_vector_type(4))) float v4f;
typedef __attribute__((ext_vector_type(8))) float v8f;
typedef __attribute__((ext_vector_type(4))) int   v4i;

#if __has_builtin(__builtin_amdgcn_global_load_async_to_lds_b128) && \
    __has_builtin(__builtin_amdgcn_s_wait_asynccnt)
#define BIAFFINE_ASYNC_LDS 1
typedef __attribute__((address_space(3))) v4i lds_v4i;
#else
#define BIAFFINE_ASYNC_LDS 0
#endif

namespace {

constexpr int kB = 8;
constexpr int kS = 1024;
constexpr int kD = 1024;
constexpr int TILE = 128;   // macro-tile edge (M and N)
constexpr int KCHUNK = 32;  // K per LDS stage
constexpr int LDSS = 36;    // padded LDS row stride in floats (144B: 16B aligned, conflict-free)

// ---- U transpose: Ut[k][d] = U[d][k] (so stage-1 becomes A @ B^T form) ----
__global__ __launch_bounds__(256)
void biaffine_transpose_kernel(const float* __restrict__ U, float* __restrict__ Ut) {
    __shared__ float t[32][33];
    const int bx = blockIdx.x * 32, by = blockIdx.y * 32;
    const int x = threadIdx.x & 31;
    const int y = threadIdx.x >> 5;  // 0..7
#pragma unroll
    for (int yy = 0; yy < 32; yy += 8)
        t[y + yy][x] = U[(size_t)(by + y + yy) * kD + bx + x];
    __syncthreads();
#pragma unroll
    for (int yy = 0; yy < 32; yy += 8)
        Ut[(size_t)(bx + y + yy) * kD + by + x] = t[x][y + yy];
}

// ---- per-row dot product: out[row] = X[row,:] . w[:]  (one wave32 per row) ----
__global__ __launch_bounds__(256)
void biaffine_rowdot_kernel(const float* __restrict__ X, const float* __restrict__ w,
                            float* __restrict__ out) {
    const int lane = threadIdx.x & 31;
    const int wv = threadIdx.x >> 5;
    const size_t row = (size_t)blockIdx.x * 8 + wv;
    const float* x = X + row * kD;
    float s = 0.0f;
#pragma unroll 4
    for (int c = lane; c < kD; c += 32) s += x[c] * w[c];
#pragma unroll
    for (int off = 16; off > 0; off >>= 1) s += __shfl_xor(s, off, 32);
    if (lane == 0) out[row] = s;
}

// ---- C = A @ B^T via V_WMMA_F32_16X16X4_F32, optional fused epilogue ----
// A: [M x K] row-major, B: [N x K] row-major, C: [M x N] row-major.
// If sh != nullptr: C += sh[bz*kS + i] + sd[bz*kS + j] + bptr[0].
__global__ __launch_bounds__(256)
void biaffine_wmma_gemm_abt_kernel(const float* __restrict__ Abase,
                                   const float* __restrict__ Bbase,
                                   float* __restrict__ Cbase,
                                   int M, int N, int K,
                                   long strideA, long strideB, long strideC,
                                   const float* __restrict__ sh,
                                   const float* __restrict__ sd,
                                   const float* __restrict__ bptr) {
    __shared__ float lA[2][TILE * LDSS];
    __shared__ float lB[2][TILE * LDSS];

    const int bz = blockIdx.z;
    const float* A = Abase + (size_t)bz * (size_t)strideA;
    const float* Bm = Bbase + (size_t)bz * (size_t)strideB;
    float* C = Cbase + (size_t)bz * (size_t)strideC;

    const int i0 = blockIdx.x * TILE;
    const int j0 = blockIdx.y * TILE;
    const int tid = threadIdx.x;
    const int lane = tid & 31;
    const int w = tid >> 5;     // wave 0..7
    const int wi = w & 3;       // i strip of 32
    const int wj = w >> 2;      // j strip of 64
    const int lr = lane & 15;          // fragment row (A) / col-row (B)
    const int lk = (lane >> 4) << 1;   // k sub-offset: 0 or 2

    // staging map: thread t owns float4s at row = t>>3 (+32q), col = (t&7)*4
    const int srow = tid >> 3;
    const int scol = (tid & 7) << 2;
    const int lbase = srow * LDSS + scol;
    const float* gA0 = A + (size_t)(i0 + srow) * (size_t)K + scol;
    const float* gB0 = Bm + (size_t)(j0 + srow) * (size_t)K + scol;

    v8f acc[2][4];
    const v8f vzero = {};
#pragma unroll
    for (int a = 0; a < 2; ++a)
#pragma unroll
        for (int c = 0; c < 4; ++c) acc[a][c] = vzero;

    auto compute_chunk = [&](const float* bufA, const float* bufB) {
#pragma unroll
        for (int kk = 0; kk < KCHUNK; kk += 4) {
            v2f af[2], bf[4];
#pragma unroll
            for (int ti = 0; ti < 2; ++ti)
                af[ti] = *(const v2f*)(bufA + (wi * 32 + ti * 16 + lr) * LDSS + kk + lk);
#pragma unroll
            for (int tj = 0; tj < 4; ++tj)
                bf[tj] = *(const v2f*)(bufB + (wj * 64 + tj * 16 + lr) * LDSS + kk + lk);
#pragma unroll
            for (int ti = 0; ti < 2; ++ti)
#pragma unroll
                for (int tj = 0; tj < 4; ++tj)
                    acc[ti][tj] = __builtin_amdgcn_wmma_f32_16x16x4_f32(
                        false, af[ti], false, bf[tj],
                        (short)0, acc[ti][tj], false, false);
        }
    };

#if BIAFFINE_ASYNC_LDS
    // ---- double-buffered async global->LDS pipeline (ASYNCcnt) ----
    auto stage_async = [&](int buf, int k0) {
#pragma unroll
        for (int q = 0; q < 4; ++q) {
            __builtin_amdgcn_global_load_async_to_lds_b128(
                (v4i*)(gA0 + k0 + (size_t)q * 32 * (size_t)K),
                (lds_v4i*)(&lA[buf][lbase + q * 32 * LDSS]), 0, 0);
            __builtin_amdgcn_global_load_async_to_lds_b128(
                (v4i*)(gB0 + k0 + (size_t)q * 32 * (size_t)K),
                (lds_v4i*)(&lB[buf][lbase + q * 32 * LDSS]), 0, 0);
        }
    };

    stage_async(0, 0);
    const int nch = K / KCHUNK;
    for (int c = 0; c < nch; ++c) {
        const int cur = c & 1;
        __builtin_amdgcn_s_wait_asynccnt(0);   // this wave's copies for chunk c done
        __syncthreads();                       // all waves' copies visible; prev reads retired
        if (c + 1 < nch) stage_async(cur ^ 1, (c + 1) * KCHUNK);
        compute_chunk(lA[cur], lB[cur]);
    }
#else
    // ---- fallback: synchronous staging, loads batched before LDS stores ----
    for (int k0 = 0; k0 < K; k0 += KCHUNK) {
        v4f ta[4], tb[4];
#pragma unroll
        for (int q = 0; q < 4; ++q) {
            ta[q] = *(const v4f*)(gA0 + k0 + (size_t)q * 32 * (size_t)K);
            tb[q] = *(const v4f*)(gB0 + k0 + (size_t)q * 32 * (size_t)K);
        }
#pragma unroll
        for (int q = 0; q < 4; ++q) {
            *(v4f*)(&lA[0][lbase + q * 32 * LDSS]) = ta[q];
            *(v4f*)(&lB[0][lbase + q * 32 * LDSS]) = tb[q];
        }
        __syncthreads();
        compute_chunk(lA[0], lB[0]);
        __syncthreads();
    }
#endif

    // writeback; C/D layout: N = lane%16, M = vgpr_r + 8*(lane/16)
    const bool fused = (sh != nullptr);
    const float bias = fused ? bptr[0] : 0.0f;
#pragma unroll
    for (int ti = 0; ti < 2; ++ti) {
        const int rbase = i0 + wi * 32 + ti * 16 + ((lane >> 4) << 3);
#pragma unroll
        for (int tj = 0; tj < 4; ++tj) {
            const int col = j0 + wj * 64 + tj * 16 + lr;
            float addc = 0.0f;
            if (fused) addc = sd[(size_t)bz * kS + col] + bias;
#pragma unroll
            for (int r = 0; r < 8; ++r) {
                float v = acc[ti][tj][r] + addc;
                if (fused) v += sh[(size_t)bz * kS + rbase + r];
                C[(size_t)(rbase + r) * (size_t)N + col] = v;
            }
        }
    }
}

}  // namespace

extern "C" void kernel_launch(void* const* d_in, const int* in_sizes, int n_in,
                              void* d_out, int out_size, void* d_ws, size_t ws_size,
                              hipStream_t stream) {
    (void)in_sizes; (void)n_in; (void)out_size; (void)ws_size;

    const float* head   = (const float*)d_in[0];  // [B,S,D]
    const float* dep    = (const float*)d_in[1];  // [B,S,D]
    const float* edge_U = (const float*)d_in[2];  // [D,D]
    const float* edge_W = (const float*)d_in[3];  // [1,2D]
    const float* edge_b = (const float*)d_in[4];  // [1]
    float* out = (float*)d_out;                   // [B,S,S]

    float* ws  = (float*)d_ws;
    float* Ut  = ws;                              // D*D        (4 MB)
    float* Mws = Ut + (size_t)kD * kD;            // B*S*D      (32 MB)
    float* sh  = Mws + (size_t)kB * kS * kD;      // B*S
    float* sd  = sh + (size_t)kB * kS;            // B*S

    // 0) Ut = U^T
    biaffine_transpose_kernel<<<dim3(kD / 32, kD / 32), 256, 0, stream>>>(edge_U, Ut);
    // 1) s_head / s_dep projections
    biaffine_rowdot_kernel<<<dim3(kB * kS / 8), 256, 0, stream>>>(head, edge_W, sh);
    biaffine_rowdot_kernel<<<dim3(kB * kS / 8), 256, 0, stream>>>(dep, edge_W + kD, sd);
    // 2) M = head @ U  ==  head @ Ut^T   (rows = B*S)
    biaffine_wmma_gemm_abt_kernel<<<dim3(kB * kS / TILE, kD / TILE, 1), 256, 0, stream>>>(
        head, Ut, Mws, kB * kS, kD, kD, 0, 0, 0, nullptr, nullptr, nullptr);
    // 3) out[b] = M[b] @ dep[b]^T + sh[b,i] + sd[b,j] + bias
    biaffine_wmma_gemm_abt_kernel<<<dim3(kS / TILE, kS / TILE, kB), 256, 0, stream>>>(
        Mws, dep, out, kS, kS, kD,
        (long)kS * kD, (long)kS * kD, (long)kS * kS, sh, sd, edge_b);
}